// MPSGNN_58987080843872
// MI455X (gfx1250) — compile-verified
//
#include <hip/hip_runtime.h>
#include <hip/hip_bf16.h>

typedef __attribute__((ext_vector_type(2))) float v2f;
typedef __attribute__((ext_vector_type(8))) float v8f;

#define DD 128
#define SCAN_B 256

// ---------------------------------------------------------------- zero (uint)
__global__ void zero_u(unsigned* __restrict__ p, int n) {
    int i = blockIdx.x * blockDim.x + threadIdx.x;
    if (i < n) p[i] = 0u;
}

// ---------------------------------------------------------------- in-degree count
__global__ void count_edges(const int* __restrict__ dst, unsigned* __restrict__ cntU,
                            int nedges) {
    int e = blockIdx.x * blockDim.x + threadIdx.x;
    if (e < nedges) atomicAdd(&cntU[dst[e]], 1u);
}

// ---------------------------------------------------------------- exclusive scan, 3-kernel hierarchy
__global__ void scan1(const unsigned* __restrict__ in, unsigned* __restrict__ out,
                      unsigned* __restrict__ bsum, int n) {
    __shared__ unsigned tmp[SCAN_B];
    int i = blockIdx.x * SCAN_B + threadIdx.x;
    unsigned v = (i < n) ? in[i] : 0u;
    tmp[threadIdx.x] = v;
    __syncthreads();
    for (int d = 1; d < SCAN_B; d <<= 1) {
        unsigned t = (threadIdx.x >= d) ? tmp[threadIdx.x - d] : 0u;
        __syncthreads();
        tmp[threadIdx.x] += t;
        __syncthreads();
    }
    unsigned incl = tmp[threadIdx.x];
    if (i < n) out[i] = incl - v;                       // exclusive
    if (threadIdx.x == SCAN_B - 1) bsum[blockIdx.x] = incl;
}

__global__ void scan2(unsigned* __restrict__ bsum, int nb) {   // 1 block, 512 threads
    __shared__ unsigned tmp[512];
    int t = threadIdx.x;
    unsigned v = (t < nb) ? bsum[t] : 0u;
    tmp[t] = v;
    __syncthreads();
    for (int d = 1; d < 512; d <<= 1) {
        unsigned x = (t >= d) ? tmp[t - d] : 0u;
        __syncthreads();
        tmp[t] += x;
        __syncthreads();
    }
    if (t < nb) bsum[t] = tmp[t] - v;                   // exclusive block offsets
}

__global__ void scan3(unsigned* __restrict__ off, unsigned* __restrict__ cursor,
                      const unsigned* __restrict__ bsum, int n) {
    int i = blockIdx.x * SCAN_B + threadIdx.x;
    if (i < n) {
        unsigned o = off[i] + bsum[blockIdx.x];
        off[i] = o;
        cursor[i] = o;
    }
}

// ---------------------------------------------------------------- CSR adjacency fill
__global__ void fill_csr(const int* __restrict__ src, const int* __restrict__ dst,
                         unsigned* __restrict__ cursor, unsigned* __restrict__ eidx,
                         int nedges) {
    int e = blockIdx.x * blockDim.x + threadIdx.x;
    if (e < nedges) {
        unsigned pos = atomicAdd(&cursor[dst[e]], 1u);
        eidx[pos] = (unsigned)src[e];
    }
}

// ---------------------------------------------------------------- pull-style gather (no fp atomics)
// one wave32 per dst node; lane owns a float4 slice of the 128-wide row.
__global__ void __launch_bounds__(256)
gather_rows(const float* __restrict__ xsrc, const unsigned* __restrict__ off,
            const unsigned* __restrict__ cntU, const unsigned* __restrict__ eidx,
            float* __restrict__ agg, int n) {
    int wave = threadIdx.x >> 5;
    int lane = threadIdx.x & 31;
    int node = blockIdx.x * 8 + wave;
    if (node >= n) return;
    unsigned start = off[node];
    unsigned deg   = cntU[node];
    float4 acc = make_float4(0.f, 0.f, 0.f, 0.f);
    unsigned j = 0;
    for (; j + 2 <= deg; j += 2) {                      // unroll 2: overlap index+row loads
        unsigned s0 = eidx[start + j];
        unsigned s1 = eidx[start + j + 1];
        float4 x0 = ((const float4*)(xsrc + (size_t)s0 * DD))[lane];
        float4 x1 = ((const float4*)(xsrc + (size_t)s1 * DD))[lane];
        acc.x += x0.x; acc.y += x0.y; acc.z += x0.z; acc.w += x0.w;
        acc.x += x1.x; acc.y += x1.y; acc.z += x1.z; acc.w += x1.w;
    }
    if (j < deg) {
        unsigned s0 = eidx[start + j];
        float4 x0 = ((const float4*)(xsrc + (size_t)s0 * DD))[lane];
        acc.x += x0.x; acc.y += x0.y; acc.z += x0.z; acc.w += x0.w;
    }
    ((float4*)(agg + (size_t)node * DD))[lane] = acc;
}

// ---------------------------------------------------------------- fused SAGE layer GEMM (fp32 WMMA)
// out[n][m] = act( (agg[n]/max(cnt[n],1)) @ Wl.T + bl + xdst[n] @ Wr.T )
__global__ void __launch_bounds__(256)
sage_gemm(const float* __restrict__ agg, const unsigned* __restrict__ cntU,
          const float* __restrict__ xdst,
          const float* __restrict__ Wl, const float* __restrict__ bl,
          const float* __restrict__ Wr,
          float* __restrict__ out, int nrows, int relu) {
    int lane = threadIdx.x & 31;
    int wv   = threadIdx.x >> 5;
    int half = lane >> 4;
    int idx  = lane & 15;
    int row  = blockIdx.x * 16 + idx;
    int rldr = row < nrows ? row : (nrows - 1);         // clamp: keep EXEC all-ones
    int col  = wv * 16 + idx;

    float inv = 1.0f / fmaxf((float)cntU[rldr], 1.0f);

    const float* arow = agg  + (size_t)rldr * DD + 2 * half;
    const float* xrow = xdst + (size_t)rldr * DD + 2 * half;
    const float* wlp  = Wl   + (size_t)col  * DD + 2 * half;   // B[k][n] = Wl[n][k]
    const float* wrp  = Wr   + (size_t)col  * DD + 2 * half;

    v8f c = {};
#pragma unroll
    for (int k = 0; k < DD; k += 4) {
        v2f a = *(const v2f*)(arow + k);
        a *= inv;
        v2f b = *(const v2f*)(wlp + k);
        c = __builtin_amdgcn_wmma_f32_16x16x4_f32(false, a, false, b,
                                                  (short)0, c, false, false);
    }
#pragma unroll
    for (int k = 0; k < DD; k += 4) {
        v2f a = *(const v2f*)(xrow + k);
        v2f b = *(const v2f*)(wrp + k);
        c = __builtin_amdgcn_wmma_f32_16x16x4_f32(false, a, false, b,
                                                  (short)0, c, false, false);
    }

    float bias = bl[col];
    int rbase = blockIdx.x * 16 + 8 * half;             // C layout: M = v + 8*half
#pragma unroll
    for (int v = 0; v < 8; ++v) {
        int r = rbase + v;
        if (r < nrows) {
            float x = c[v] + bias;
            if (relu) x = fmaxf(x, 0.0f);
            out[(size_t)r * DD + col] = x;
        }
    }
}

// ---------------------------------------------------------------- fold outW/reg:  v_p = outW_p.T @ regW_p
__global__ void prep_v(const float* __restrict__ outW0, const float* __restrict__ outb0,
                       const float* __restrict__ outW1, const float* __restrict__ outb1,
                       const float* __restrict__ regW, const float* __restrict__ regb,
                       float* __restrict__ v) {
    int k = threadIdx.x;   // 128 threads, 1 block
    float s0 = 0.0f, s1 = 0.0f;
    for (int m = 0; m < DD; ++m) {
        s0 += outW0[m * DD + k] * regW[m];
        s1 += outW1[m * DD + k] * regW[DD + m];
    }
    v[k]      = s0;
    v[DD + k] = s1;
    if (k == 0) {
        float cc = regb[0];
        for (int m = 0; m < DD; ++m)
            cc += outb0[m] * regW[m] + outb1[m] * regW[DD + m];
        v[2 * DD] = cc;
    }
}

// ---------------------------------------------------------------- final: out[n] = c + hu0[n].v0 + hu1[n].v1
__global__ void __launch_bounds__(256)
final_reg(const float* __restrict__ hu0, const float* __restrict__ hu1,
          const float* __restrict__ v, float* __restrict__ out, int n) {
    int wave = threadIdx.x >> 5;
    int lane = threadIdx.x & 31;
    int node = blockIdx.x * 8 + wave;
    if (node >= n) return;

    float4 a  = ((const float4*)(hu0 + (size_t)node * DD))[lane];
    float4 w0 = ((const float4*)(v))[lane];
    float acc = a.x * w0.x + a.y * w0.y + a.z * w0.z + a.w * w0.w;
    float4 b  = ((const float4*)(hu1 + (size_t)node * DD))[lane];
    float4 w1 = ((const float4*)(v + DD))[lane];
    acc += b.x * w1.x + b.y * w1.y + b.z * w1.z + b.w * w1.w;

    for (int off = 16; off > 0; off >>= 1)
        acc += __shfl_down(acc, off, 32);
    if (lane == 0) out[node] = acc + v[2 * DD];
}

// ----------------------------------------------------------------
struct HopBufs {
    unsigned *cntU, *off, *cursor, *eidx, *bsum;
    float* agg;
};

static void run_hop(const float* xsrc, const int* src, const int* dst,
                    const float* xdst, const float* Wl, const float* bl,
                    const float* Wr, float* outbuf,
                    int N, int E, const HopBufs& b, hipStream_t stream) {
    const int nbN = (N + SCAN_B - 1) / SCAN_B;
    zero_u<<<dim3((N + 255) / 256), 256, 0, stream>>>(b.cntU, N);
    count_edges<<<dim3((E + 255) / 256), 256, 0, stream>>>(dst, b.cntU, E);
    scan1<<<dim3(nbN), SCAN_B, 0, stream>>>(b.cntU, b.off, b.bsum, N);
    scan2<<<dim3(1), 512, 0, stream>>>(b.bsum, nbN);
    scan3<<<dim3(nbN), SCAN_B, 0, stream>>>(b.off, b.cursor, b.bsum, N);
    fill_csr<<<dim3((E + 255) / 256), 256, 0, stream>>>(src, dst, b.cursor, b.eidx, E);
    gather_rows<<<dim3((N + 7) / 8), 256, 0, stream>>>(xsrc, b.off, b.cntU, b.eidx, b.agg, N);
    sage_gemm<<<dim3((N + 15) / 16), 256, 0, stream>>>(b.agg, b.cntU, xdst, Wl, bl, Wr,
                                                       outbuf, N, 1);
}

extern "C" void kernel_launch(void* const* d_in, const int* in_sizes, int n_in,
                              void* d_out, int out_size, void* d_ws, size_t ws_size,
                              hipStream_t stream) {
    const float* x_user = (const float*)d_in[0];
    const float* x_item = (const float*)d_in[1];
    const int*   ei00   = (const int*)d_in[2];
    const int*   ei01   = (const int*)d_in[3];
    const int*   ei10   = (const int*)d_in[4];
    const int*   ei11   = (const int*)d_in[5];
    const float* Wl00 = (const float*)d_in[6];
    const float* Wr00 = (const float*)d_in[7];
    const float* Wl01 = (const float*)d_in[8];
    const float* Wr01 = (const float*)d_in[9];
    const float* Wl10 = (const float*)d_in[10];
    const float* Wr10 = (const float*)d_in[11];
    const float* Wl11 = (const float*)d_in[12];
    const float* Wr11 = (const float*)d_in[13];
    const float* outW0 = (const float*)d_in[14];
    const float* outW1 = (const float*)d_in[15];
    const float* bl00 = (const float*)d_in[16];
    const float* bl01 = (const float*)d_in[17];
    const float* bl10 = (const float*)d_in[18];
    const float* bl11 = (const float*)d_in[19];
    const float* outb0 = (const float*)d_in[20];
    const float* outb1 = (const float*)d_in[21];
    const float* regW  = (const float*)d_in[22];
    const float* regb  = (const float*)d_in[23];

    const int N = in_sizes[0] / DD;
    const int E = in_sizes[2] / 2;
    const size_t ND = (size_t)N * DD;

    // ---- workspace layout (float region first, 512B-aligned rows) ----
    float* wsf = (float*)d_ws;
    float* agg = wsf;                 // N*D
    float* hi  = agg + ND;            // N*D
    float* hu0 = hi + ND;             // N*D
    float* hu1 = hu0 + ND;            // N*D
    float* v   = hu1 + ND;            // 260 (16B aligned)
    unsigned* wsU    = (unsigned*)(v + 260);
    unsigned* cntU   = wsU;           // N
    unsigned* off    = cntU + N;      // N
    unsigned* cursor = off + N;       // N
    unsigned* eidx   = cursor + N;    // E
    unsigned* bsum   = eidx + E;      // 512
    float* out = (float*)d_out;

    HopBufs hb{cntU, off, cursor, eidx, bsum, agg};

    // ---- metapath 0: user -> item -> user
    run_hop(x_user, ei00, ei00 + E, x_item, Wl00, bl00, Wr00, hi,  N, E, hb, stream);
    run_hop(hi,     ei01, ei01 + E, x_user, Wl01, bl01, Wr01, hu0, N, E, hb, stream);
    // ---- metapath 1
    run_hop(x_user, ei10, ei10 + E, x_item, Wl10, bl10, Wr10, hi,  N, E, hb, stream);
    run_hop(hi,     ei11, ei11 + E, x_user, Wl11, bl11, Wr11, hu1, N, E, hb, stream);

    // ---- folded outW/reg stage
    prep_v<<<dim3(1), 128, 0, stream>>>(outW0, outb0, outW1, outb1, regW, regb, v);
    final_reg<<<dim3((N + 7) / 8), 256, 0, stream>>>(hu0, hu1, v, out, N);
}